// Net_53446573032186
// MI455X (gfx1250) — compile-verified
//
#include <hip/hip_runtime.h>
#include <hip/hip_bf16.h>

// ---------------------------------------------------------------------------
// 2-layer GCN for MI455X (gfx1250), CSR-gather formulation.
//   Layer1: h = relu( b1 + D^-1/2 A D^-1/2 (x@W1) )   (A incl. self-loops)
//   Layer2: out = sigmoid( b4 + D^-1/2 A D^-1/2 (h@W4) )
//
// Roofline: GEMM1 is 4.2 GFLOP (f32 WMMA, trivial). Aggregation working set
// (xw 51MB + h 51MB + CSR 6.4MB) fits in the 192MB L2, so the edge gather
// (~820MB of L2 reads) runs in tens of microseconds — vs ~205M serialized
// f32 atomics in the scatter form. Only int atomics remain (histogram +
// bucket tickets, 3.2M total).
// ---------------------------------------------------------------------------

typedef __attribute__((ext_vector_type(2))) float v2f;
typedef __attribute__((ext_vector_type(8))) float v8f;

#define FIN   165   // input feature dim
#define FPAD  168   // padded to multiple of 4 (f32 WMMA K-step)
#define FOUT  128   // hidden dim

// ---------------- histogram / normalization ----------------

__global__ void k_zero_int(int* __restrict__ p, int n) {
  int i = blockIdx.x * blockDim.x + threadIdx.x;
  if (i < n) p[i] = 0;
}

__global__ void k_count(const int* __restrict__ dst, int E, int* __restrict__ cnt) {
  int i = blockIdx.x * blockDim.x + threadIdx.x;
  if (i < E) atomicAdd(&cnt[dst[i]], 1);
}

// deg includes +1 for the self-loop, so it is always >= 1.
__global__ void k_dinv(const int* __restrict__ cnt, float* __restrict__ dinv, int n) {
  int i = blockIdx.x * blockDim.x + threadIdx.x;
  if (i < n) dinv[i] = rsqrtf((float)cnt[i] + 1.0f);
}

// ---------------- exclusive scan (3 small kernels) ----------------

// per-256-chunk sums
__global__ __launch_bounds__(256) void k_bsum(const int* __restrict__ cnt,
                                              int* __restrict__ bsum, int n) {
  __shared__ int sdata[8];
  int i = blockIdx.x * 256 + threadIdx.x;
  int v = (i < n) ? cnt[i] : 0;
  for (int off = 16; off > 0; off >>= 1) v += __shfl_xor(v, off, 32);
  if ((threadIdx.x & 31) == 0) sdata[threadIdx.x >> 5] = v;
  __syncthreads();
  if (threadIdx.x == 0) {
    int t = 0;
    for (int j = 0; j < 8; ++j) t += sdata[j];
    bsum[blockIdx.x] = t;
  }
}

// single-block exclusive scan of chunk sums (in place), chunked with carry
__global__ __launch_bounds__(512) void k_scan_bsum(int* __restrict__ bsum, int nb) {
  __shared__ int tmp[512];
  __shared__ int carry;
  int tid = threadIdx.x;
  if (tid == 0) carry = 0;
  __syncthreads();
  for (int base = 0; base < nb; base += 512) {
    int i = base + tid;
    int v = (i < nb) ? bsum[i] : 0;
    tmp[tid] = v;
    __syncthreads();
    for (int off = 1; off < 512; off <<= 1) {
      int t = (tid >= off) ? tmp[tid - off] : 0;
      __syncthreads();
      tmp[tid] += t;
      __syncthreads();
    }
    if (i < nb) bsum[i] = carry + tmp[tid] - v;   // exclusive
    __syncthreads();
    if (tid == 511) carry += tmp[511];
    __syncthreads();
  }
}

// per-chunk rescan + chunk prefix -> rowptr (and write cursor copy)
__global__ __launch_bounds__(256) void k_rowptr(const int* __restrict__ cnt,
                                                const int* __restrict__ bpre,
                                                int* __restrict__ rowptr,
                                                int* __restrict__ wcur, int n) {
  __shared__ int tmp[256];
  int tid = threadIdx.x;
  int i = blockIdx.x * 256 + tid;
  int v = (i < n) ? cnt[i] : 0;
  tmp[tid] = v;
  __syncthreads();
  for (int off = 1; off < 256; off <<= 1) {
    int t = (tid >= off) ? tmp[tid - off] : 0;
    __syncthreads();
    tmp[tid] += t;
    __syncthreads();
  }
  if (i < n) {
    int e = bpre[blockIdx.x] + tmp[tid] - v;
    rowptr[i] = e;
    wcur[i]   = e;
    if (i == n - 1) rowptr[n] = e + v;
  }
}

// counting-sort edges by dst: esrc[bucket(dst)] = src
__global__ void k_fill(const int* __restrict__ src, const int* __restrict__ dst, int E,
                       int* __restrict__ wcur, int* __restrict__ esrc) {
  int e = blockIdx.x * blockDim.x + threadIdx.x;
  if (e < E) {
    int pos = atomicAdd(&wcur[dst[e]], 1);
    esrc[pos] = src[e];
  }
}

// ---------------- WMMA GEMM: xw[n,128] = x[n,165] @ W1[165,128] -------------
// Block = 256 threads = 8 waves; each wave owns one 16x16 N-tile. x tile
// staged in LDS (zero-padded to 168); W1 (84.5KB) stays L2/WGP$-resident.
__global__ __launch_bounds__(256) void k_gemm_wmma(const float* __restrict__ x,
                                                   const float* __restrict__ W1,
                                                   float* __restrict__ xw,
                                                   int n) {
  __shared__ float xs[16][FPAD];

  const int row0 = blockIdx.x * 16;
  const int tid  = threadIdx.x;

  for (int idx = tid; idx < 16 * FPAD; idx += 256) {
    int r = idx / FPAD, c = idx % FPAD;
    float v = 0.0f;
    int row = row0 + r;
    if (c < FIN && row < n) v = x[(size_t)row * FIN + c];
    xs[r][c] = v;
  }
  __syncthreads();

  const int wave  = tid >> 5;
  const int lane  = tid & 31;
  const int n0    = wave * 16;
  const int m     = lane & 15;
  const int khalf = lane >> 4;

  v8f acc = {};
  for (int ks = 0; ks < FPAD / 4; ++ks) {
    const int kb = ks * 4 + 2 * khalf;
    v2f a;                                  // A 16x4: lane m -> row m, K pair
    a.x = xs[m][kb];
    a.y = xs[m][kb + 1];
    const int col = n0 + m;
    v2f b;                                  // B 4x16: lane -> col, K pair
    b.x = (kb     < FIN) ? W1[(size_t)kb       * FOUT + col] : 0.0f;
    b.y = (kb + 1 < FIN) ? W1[(size_t)(kb + 1) * FOUT + col] : 0.0f;
    acc = __builtin_amdgcn_wmma_f32_16x16x4_f32(false, a, false, b,
                                                (short)0, acc, false, false);
  }

  const int ccol  = n0 + m;
  const int rbase = row0 + 8 * khalf;
  for (int r = 0; r < 8; ++r) {
    int row = rbase + r;
    if (row < n) xw[(size_t)row * FOUT + ccol] = acc[r];
  }
}

// ---------------- layer-1 gather: h = relu(b1 + dinv_i*(dinv_i*xw_i + sum))
// One wave32 per dst node; lane -> 4 cols (float4). No atomics: reads the
// CSR bucket, coalesced 512B row gathers from L2-resident xw.
__global__ __launch_bounds__(256) void k_gather1(const float* __restrict__ xw,
                                                 const int* __restrict__ rowptr,
                                                 const int* __restrict__ esrc,
                                                 const float* __restrict__ dinv,
                                                 const float* __restrict__ b1,
                                                 float* __restrict__ h, int n) {
  int node = blockIdx.x * 8 + (threadIdx.x >> 5);
  int lane = threadIdx.x & 31;
  if (node >= n) return;
  int c0 = lane * 4;
  float di = dinv[node];

  const float4 xself = *(const float4*)(xw + (size_t)node * FOUT + c0);
  float4 acc = make_float4(di * xself.x, di * xself.y, di * xself.z, di * xself.w);

  int beg = rowptr[node], end = rowptr[node + 1];
  for (int e = beg; e < end; ++e) {
    int sidx = esrc[e];
    float w  = dinv[sidx];
    const float4 v = *(const float4*)(xw + (size_t)sidx * FOUT + c0);
    acc.x += w * v.x; acc.y += w * v.y; acc.z += w * v.z; acc.w += w * v.w;
  }

  const float4 bb = *(const float4*)(b1 + c0);
  float4 r;
  r.x = fmaxf(bb.x + di * acc.x, 0.0f);
  r.y = fmaxf(bb.y + di * acc.y, 0.0f);
  r.z = fmaxf(bb.z + di * acc.z, 0.0f);
  r.w = fmaxf(bb.w + di * acc.w, 0.0f);
  *(float4*)(h + (size_t)node * FOUT + c0) = r;
}

// s[i] = h[i,:] . W4   (h already relu'd) — wave32 shuffle reduce
__global__ __launch_bounds__(256) void k_dot(const float* __restrict__ h,
                                             const float* __restrict__ W4,
                                             float* __restrict__ s, int n) {
  int node = blockIdx.x * 8 + (threadIdx.x >> 5);
  int lane = threadIdx.x & 31;
  if (node >= n) return;
  const float* hp = h + (size_t)node * FOUT + lane * 4;
  float acc = 0.0f;
#pragma unroll
  for (int j = 0; j < 4; ++j) acc += hp[j] * W4[lane * 4 + j];
  for (int off = 16; off > 0; off >>= 1) acc += __shfl_xor(acc, off, 32);
  if (lane == 0) s[node] = acc;
}

// layer-2 gather + bias + sigmoid, one thread per node (scalar messages)
__global__ void k_gather2(const float* __restrict__ s, const int* __restrict__ rowptr,
                          const int* __restrict__ esrc, const float* __restrict__ dinv,
                          const float* __restrict__ b4, float* __restrict__ out, int n) {
  int i = blockIdx.x * blockDim.x + threadIdx.x;
  if (i >= n) return;
  float di  = dinv[i];
  float acc = di * s[i];                       // self-loop
  int beg = rowptr[i], end = rowptr[i + 1];
  for (int e = beg; e < end; ++e) {
    int sidx = esrc[e];
    acc += dinv[sidx] * s[sidx];
  }
  float o = b4[0] + di * acc;
  out[i] = 1.0f / (1.0f + __expf(-o));
}

// ---------------------------------------------------------------------------

extern "C" void kernel_launch(void* const* d_in, const int* in_sizes, int n_in,
                              void* d_out, int out_size, void* d_ws, size_t ws_size,
                              hipStream_t stream) {
  const float* x  = (const float*)d_in[0];
  const int*   ei = (const int*)d_in[1];     // edge_index [2, E] (int32)
  const float* W1 = (const float*)d_in[2];
  const float* b1 = (const float*)d_in[3];
  const float* W4 = (const float*)d_in[4];
  const float* b4 = (const float*)d_in[5];
  float* out = (float*)d_out;

  const int n  = in_sizes[0] / FIN;          // 100000
  const int E  = in_sizes[1] / 2;            // 1600000
  const int nb = (n + 255) / 256;            // scan chunks
  const int* srcv = ei;
  const int* dstv = ei + E;

  // 16B-aligned workspace carve-out
  char* base = (char*)d_ws;
  size_t off = 0;
  auto carve = [&](size_t bytes) -> void* {
    off = (off + 15) & ~(size_t)15;
    void* p = base + off;
    off += bytes;
    return p;
  };
  int*   cnt    = (int*)carve((size_t)n * 4);
  int*   rowptr = (int*)carve(((size_t)n + 1) * 4);
  int*   wcur   = (int*)carve((size_t)n * 4);
  int*   bsum   = (int*)carve((size_t)nb * 4);
  int*   esrc   = (int*)carve((size_t)E * 4);
  float* dinv   = (float*)carve((size_t)n * 4);
  float* sbuf   = (float*)carve((size_t)n * 4);
  float* xw     = (float*)carve((size_t)n * FOUT * 4);
  float* h      = (float*)carve((size_t)n * FOUT * 4);

  const int B = 256;
  // degree histogram (self-init each call for determinism)
  k_zero_int<<<(n + B - 1) / B, B, 0, stream>>>(cnt, n);
  k_count<<<(E + B - 1) / B, B, 0, stream>>>(dstv, E, cnt);
  k_dinv<<<(n + B - 1) / B, B, 0, stream>>>(cnt, dinv, n);
  // CSR build: scan + counting sort by dst
  k_bsum<<<nb, 256, 0, stream>>>(cnt, bsum, n);
  k_scan_bsum<<<1, 512, 0, stream>>>(bsum, nb);
  k_rowptr<<<nb, 256, 0, stream>>>(cnt, bsum, rowptr, wcur, n);
  k_fill<<<(E + B - 1) / B, B, 0, stream>>>(srcv, dstv, E, wcur, esrc);
  // dense GEMM on the matrix pipe
  k_gemm_wmma<<<(n + 15) / 16, 256, 0, stream>>>(x, W1, xw, n);
  // layer-1 aggregation (gather, no atomics) + bias + relu
  k_gather1<<<(n + 7) / 8, 256, 0, stream>>>(xw, rowptr, esrc, dinv, b1, h, n);
  // h @ W4
  k_dot<<<(n + 7) / 8, 256, 0, stream>>>(h, W4, sbuf, n);
  // layer-2 aggregation + bias + sigmoid
  k_gather2<<<(n + B - 1) / B, B, 0, stream>>>(sbuf, rowptr, esrc, dinv, b4, out, n);
}